// GCNModel_72945724555834
// MI455X (gfx1250) — compile-verified
//
#include <hip/hip_runtime.h>
#include <hip/hip_bf16.h>

typedef __attribute__((ext_vector_type(2))) float v2f;
typedef __attribute__((ext_vector_type(8))) float v8f;

#define N_USER 100000
#define N_ITEM 150000
#define N_NODES (N_USER + N_ITEM)
#define N_HID 64
#define E_UI 1000000
#define E_S 1000000
#define SLOPE 0.5f
#define EPS 1e-12f

// ---------------------------------------------------------------------------
// copy src -> X and src -> out (residual stream init)
// ---------------------------------------------------------------------------
__global__ void copy_dual_kernel(const float* __restrict__ src,
                                 float* __restrict__ X,
                                 float* __restrict__ out,
                                 int n) {
    int i = blockIdx.x * blockDim.x + threadIdx.x;
    if (i < n) {
        float v = src[i];
        X[i] = v;
        out[i] = v;
    }
}

// ---------------------------------------------------------------------------
// zero fill
// ---------------------------------------------------------------------------
__global__ void zero_kernel(float* __restrict__ p, int n) {
    int i = blockIdx.x * blockDim.x + threadIdx.x;
    if (i < n) p[i] = 0.0f;
}

// ---------------------------------------------------------------------------
// Y[M,64] = X[M,64] @ W[64,64] with V_WMMA_F32_16X16X4_F32.
// One wave per 16-row tile; 4 column tiles (N=64), K stepped by 4.
// A frag (16x4 f32): lane (r,h): a[j] = X[row0+r][k + 2h + j]
// B frag (4x16 f32): lane (r,h): b[j] = W[k + 2h + j][ct*16 + r]
// C/D (16x16 f32):   lane (r,h): c[v] = Y[row0 + v + 8h][ct*16 + r]
// M must be a multiple of 16 (250000 and 100000 both are).
// ---------------------------------------------------------------------------
__global__ void __launch_bounds__(32)
gemm64_wmma_kernel(const float* __restrict__ X,
                   const float* __restrict__ W,
                   float* __restrict__ Y,
                   int M) {
    const int row0 = blockIdx.x * 16;
    if (row0 >= M) return;
    const int lane = threadIdx.x;
    const int r = lane & 15;
    const int h = lane >> 4;

    v8f acc[4];
    acc[0] = v8f{};
    acc[1] = v8f{};
    acc[2] = v8f{};
    acc[3] = v8f{};

#pragma unroll
    for (int k = 0; k < 64; k += 4) {
        const float* xr = X + (size_t)(row0 + r) * N_HID + k + 2 * h;
        v2f a;
        a[0] = xr[0];
        a[1] = xr[1];
        const float* w0 = W + (size_t)(k + 2 * h) * N_HID + r;       // row k+2h
        const float* w1 = w0 + N_HID;                                // row k+2h+1
#pragma unroll
        for (int ct = 0; ct < 4; ++ct) {
            v2f b;
            b[0] = w0[ct * 16];
            b[1] = w1[ct * 16];
            acc[ct] = __builtin_amdgcn_wmma_f32_16x16x4_f32(
                /*neg_a=*/false, a, /*neg_b=*/false, b,
                /*c_mod=*/(short)0, acc[ct],
                /*reuse_a=*/false, /*reuse_b=*/false);
        }
    }

#pragma unroll
    for (int ct = 0; ct < 4; ++ct) {
#pragma unroll
        for (int v = 0; v < 8; ++v) {
            Y[(size_t)(row0 + v + 8 * h) * N_HID + ct * 16 + r] = acc[ct][v];
        }
    }
}

// ---------------------------------------------------------------------------
// Z[rows[e]] += vals[e] * Y[cols[e]]   (64-wide rows, fp32 atomics)
// 4 edges per 256-thread block, 64 lanes per edge.
// ---------------------------------------------------------------------------
__global__ void __launch_bounds__(256)
spmm_scatter_kernel(const int* __restrict__ rows,
                    const int* __restrict__ cols,
                    const float* __restrict__ vals,
                    const float* __restrict__ Y,
                    float* __restrict__ Z,
                    int E) {
    int e = blockIdx.x * 4 + (threadIdx.x >> 6);
    int d = threadIdx.x & 63;
    if (e < E) {
        int rr = rows[e];
        int cc = cols[e];
        float v = vals[e];
        unsafeAtomicAdd(&Z[(size_t)rr * N_HID + d], v * Y[(size_t)cc * N_HID + d]);
    }
}

// ---------------------------------------------------------------------------
// v = leaky(Z[row]); X[row] = v (next layer input);
// out[row] += v / max(||v||, EPS)
// One wave per row; each lane owns 2 consecutive floats; shfl_xor reduce.
// ---------------------------------------------------------------------------
__global__ void __launch_bounds__(32)
leaky_norm_acc_kernel(const float* __restrict__ Z,
                      float* __restrict__ X,
                      float* __restrict__ out,
                      int M) {
    int n = blockIdx.x;
    if (n >= M) return;
    int lane = threadIdx.x;

    const float2* zr = (const float2*)(Z + (size_t)n * N_HID);
    float2 z = zr[lane];
    float2 v;
    v.x = (z.x >= 0.0f) ? z.x : SLOPE * z.x;
    v.y = (z.y >= 0.0f) ? z.y : SLOPE * z.y;

    ((float2*)(X + (size_t)n * N_HID))[lane] = v;

    float ss = v.x * v.x + v.y * v.y;
#pragma unroll
    for (int o = 16; o > 0; o >>= 1) ss += __shfl_xor(ss, o, 32);

    float inv = 1.0f / fmaxf(sqrtf(ss), EPS);

    float2* orow = (float2*)(out + (size_t)n * N_HID);
    float2 cur = orow[lane];
    cur.x += v.x * inv;
    cur.y += v.y * inv;
    orow[lane] = cur;
}

// ---------------------------------------------------------------------------
// host-side orchestration
// ---------------------------------------------------------------------------
extern "C" void kernel_launch(void* const* d_in, const int* in_sizes, int n_in,
                              void* d_out, int out_size, void* d_ws, size_t ws_size,
                              hipStream_t stream) {
    const float* user_emb = (const float*)d_in[0];
    const float* item_emb = (const float*)d_in[1];
    const float* ui_w     = (const float*)d_in[2];  // [2,64,64]
    const float* s_w      = (const float*)d_in[3];  // [2,64,64]
    const int*   ui_rows  = (const int*)d_in[4];
    const int*   ui_cols  = (const int*)d_in[5];
    const float* ui_vals  = (const float*)d_in[6];
    const int*   s_rows   = (const int*)d_in[7];
    const int*   s_cols   = (const int*)d_in[8];
    const float* s_vals   = (const float*)d_in[9];

    float* X = (float*)d_ws;                    // [N_NODES,64]
    float* Y = X + (size_t)N_NODES * N_HID;     // [N_NODES,64]
    float* Z = Y + (size_t)N_NODES * N_HID;     // [N_NODES,64]

    float* out_ui = (float*)d_out;                      // [N_NODES,64]
    float* out_s  = out_ui + (size_t)N_NODES * N_HID;   // [N_USER,64]

    const int nu = N_USER * N_HID;
    const int ni = N_ITEM * N_HID;
    const int nn = N_NODES * N_HID;

    // ---- UI path: X0 = concat(user, item); out_ui = X0 ----
    copy_dual_kernel<<<(nu + 255) / 256, 256, 0, stream>>>(user_emb, X, out_ui, nu);
    copy_dual_kernel<<<(ni + 255) / 256, 256, 0, stream>>>(item_emb, X + nu, out_ui + nu, ni);

    for (int l = 0; l < 2; ++l) {
        gemm64_wmma_kernel<<<N_NODES / 16, 32, 0, stream>>>(X, ui_w + (size_t)l * N_HID * N_HID, Y, N_NODES);
        zero_kernel<<<(nn + 255) / 256, 256, 0, stream>>>(Z, nn);
        spmm_scatter_kernel<<<(E_UI + 3) / 4, 256, 0, stream>>>(ui_rows, ui_cols, ui_vals, Y, Z, E_UI);
        leaky_norm_acc_kernel<<<N_NODES, 32, 0, stream>>>(Z, X, out_ui, N_NODES);
    }

    // ---- Social path: X = user_emb; out_s = user_emb ----
    copy_dual_kernel<<<(nu + 255) / 256, 256, 0, stream>>>(user_emb, X, out_s, nu);

    for (int l = 0; l < 2; ++l) {
        gemm64_wmma_kernel<<<N_USER / 16, 32, 0, stream>>>(X, s_w + (size_t)l * N_HID * N_HID, Y, N_USER);
        zero_kernel<<<(nu + 255) / 256, 256, 0, stream>>>(Z, nu);
        spmm_scatter_kernel<<<(E_S + 3) / 4, 256, 0, stream>>>(s_rows, s_cols, s_vals, Y, Z, E_S);
        leaky_norm_acc_kernel<<<N_USER, 32, 0, stream>>>(Z, X, out_s, N_USER);
    }
}